// QuantumKernelFeature_65481071397883
// MI455X (gfx1250) — compile-verified
//
#include <hip/hip_runtime.h>

typedef float v2f __attribute__((ext_vector_type(2)));
typedef float v8f __attribute__((ext_vector_type(8)));

#define N_DIM 8192
#define K_DIM 64
#define TILE  128
#define LDS_STRIDE 68   // 64 floats + 4 pad -> stride%64banks==4, conflict-free col reads

__launch_bounds__(256, 1)
__global__ void rbf_gram_wmma(const float* __restrict__ x, float* __restrict__ out) {
  __shared__ float As[TILE * LDS_STRIDE];
  __shared__ float Bs[TILE * LDS_STRIDE];
  __shared__ float sqA[TILE];
  __shared__ float sqB[TILE];

  const int tid  = threadIdx.x;
  const int lane = tid & 31;
  const int wave = tid >> 5;     // 0..7
  const int l    = lane & 15;    // lane within half-wave
  const int hi   = lane >> 4;    // 0 or 1 (selects K pair / M+8 half)

  const int blockRow = blockIdx.y * TILE;
  const int blockCol = blockIdx.x * TILE;

  // ---- stage 128x64 A-rows and B-rows into LDS (float4 loads; x stays L2-resident) ----
  // 128 rows * 16 float4/row = 2048 float4 per matrix; 8 per thread.
  #pragma unroll
  for (int i = 0; i < 8; ++i) {
    int idx = tid + i * 256;       // 0..2047
    int r   = idx >> 4;            // row 0..127
    int c4  = idx & 15;            // float4 column
    float4 av = ((const float4*)(x + (size_t)(blockRow + r) * K_DIM))[c4];
    float4 bv = ((const float4*)(x + (size_t)(blockCol + r) * K_DIM))[c4];
    *(float4*)(&As[r * LDS_STRIDE + c4 * 4]) = av;
    *(float4*)(&Bs[r * LDS_STRIDE + c4 * 4]) = bv;
  }
  __syncthreads();

  // ---- per-row squared norms (one row per thread) ----
  if (tid < 128) {
    float s = 0.f;
    #pragma unroll
    for (int k = 0; k < K_DIM; ++k) { float v = As[tid * LDS_STRIDE + k]; s += v * v; }
    sqA[tid] = s;
  } else {
    int r = tid - 128;
    float s = 0.f;
    #pragma unroll
    for (int k = 0; k < K_DIM; ++k) { float v = Bs[r * LDS_STRIDE + k]; s += v * v; }
    sqB[r] = s;
  }
  __syncthreads();

  // ---- fp32 WMMA: wave owns rows [16*wave, 16*wave+16), all 128 cols ----
  const int m0 = wave * 16;
  v8f acc[8] = {};

  #pragma unroll
  for (int kt = 0; kt < 16; ++kt) {
    // A frag 16x4: lanes 0-15 -> K = {4kt, 4kt+1}; lanes 16-31 -> K = {4kt+2, 4kt+3}
    const int kbase = kt * 4 + 2 * hi;
    v2f a = *(const v2f*)(&As[(m0 + l) * LDS_STRIDE + kbase]);
    #pragma unroll
    for (int nt = 0; nt < 8; ++nt) {
      // B frag 4x16 of X^T: B[k][n] = Bs[n][k] -> same (row,kpair) addressing on Bs
      v2f b = *(const v2f*)(&Bs[(nt * 16 + l) * LDS_STRIDE + kbase]);
      acc[nt] = __builtin_amdgcn_wmma_f32_16x16x4_f32(
          /*neg_a=*/false, a, /*neg_b=*/false, b,
          /*c_mod=*/(short)0, acc[nt], /*reuse_a=*/false, /*reuse_b=*/false);
    }
  }

  // ---- epilogue: d2 = |xi|^2 + |xj|^2 - 2<xi,xj>; out = exp(-0.5*d2), NT stores ----
  #pragma unroll
  for (int nt = 0; nt < 8; ++nt) {
    const float sb = sqB[nt * 16 + l];          // N = lane&15 within tile
    #pragma unroll
    for (int r = 0; r < 8; ++r) {
      const int m_local = r + 8 * hi;           // C/D layout: VGPR r holds M=r (+8 for hi half)
      const float sa = sqA[m0 + m_local];
      float d2 = sa + sb - 2.0f * acc[nt][r];
      d2 = fmaxf(d2, 0.0f);
      const float v = __expf(-0.5f * d2);       // gamma = 0.5
      const size_t row = (size_t)(blockRow + m0 + m_local);
      const size_t col = (size_t)(blockCol + nt * 16 + l);
      // 256MB single-use output: non-temporal store, don't evict L2-resident x
      __builtin_nontemporal_store(v, &out[row * N_DIM + col]);
    }
  }
}

extern "C" void kernel_launch(void* const* d_in, const int* in_sizes, int n_in,
                              void* d_out, int out_size, void* d_ws, size_t ws_size,
                              hipStream_t stream) {
  (void)in_sizes; (void)n_in; (void)d_ws; (void)ws_size; (void)out_size;
  const float* x = (const float*)d_in[0];
  float* out = (float*)d_out;
  dim3 grid(N_DIM / TILE, N_DIM / TILE);   // 64 x 64 blocks
  dim3 block(256);                         // 8 waves (wave32)
  rbf_gram_wmma<<<grid, block, 0, stream>>>(x, out);
}